// VulkanMaxPool2d_16140487098967
// MI455X (gfx1250) — compile-verified
//
#include <hip/hip_runtime.h>

// 2x2 stride-2 max pool, NCHW, VALID padding.
// Input : (16, 96, 224, 224) f32  ->  Output: (16, 96, 112, 112) f32
//
// Pure streaming kernel (zero reuse: non-overlapping windows). Optimum =
// coalesced 128-bit NT loads + NT stores, minimal index math.
// ~385 MB traffic => ~16.6 us floor at 23.3 TB/s.
//
// Index algebra: with H = 2*OH, the two input rows feeding output task
// t2 = nc*OH + oh are flattened rows 2*t2 and 2*t2+1, so nc/oh never need
// to be separated: one unsigned div by 56 yields both addresses.

typedef float vf4 __attribute__((ext_vector_type(4)));
typedef float vf2 __attribute__((ext_vector_type(2)));

#define IN_W   224u
#define OUT_W  112u
#define GW     56u   /* float4 groups per output row: OUT_W / 2 */

__global__ __launch_bounds__(256) void
maxpool2x2_s2_kernel(const float* __restrict__ x,
                     float* __restrict__ y,
                     unsigned total_tasks) {
    unsigned tid = blockIdx.x * 256u + threadIdx.x;
    if (tid >= total_tasks) return;

    // t2 = nc*112 + oh (row-pair index), g = float4 group within the row.
    // Unsigned div-by-56 => single v_mul_hi_u32 + shift, no sign fixups.
    unsigned t2 = tid / GW;
    unsigned g  = tid - t2 * GW;

    // 32-bit element offsets (input has 77M elements < 2^31):
    // keeps SGPR 64-bit base + VGPR 32-bit offset addressing.
    unsigned in_off  = t2 * (2u * IN_W) + g * 4u;   // row 2*t2, cols 4g..4g+3
    unsigned out_off = t2 * OUT_W       + g * 2u;

    // 16B-aligned, per-instruction fully coalesced (lane stride == 16B),
    // non-temporal: 385MB stream must not thrash the 192MB L2.
    vf4 r0 = __builtin_nontemporal_load((const vf4*)(x + in_off));
    vf4 r1 = __builtin_nontemporal_load((const vf4*)(x + in_off + IN_W));

    float o0 = fmaxf(fmaxf(r0.x, r0.y), fmaxf(r1.x, r1.y));
    float o1 = fmaxf(fmaxf(r0.z, r0.w), fmaxf(r1.z, r1.w));

    vf2 o;
    o.x = o0;
    o.y = o1;
    __builtin_nontemporal_store(o, (vf2*)(y + out_off));
}

extern "C" void kernel_launch(void* const* d_in, const int* in_sizes, int n_in,
                              void* d_out, int out_size, void* d_ws, size_t ws_size,
                              hipStream_t stream) {
    (void)in_sizes; (void)n_in; (void)out_size; (void)d_ws; (void)ws_size;

    const float* x = (const float*)d_in[0];
    float*       y = (float*)d_out;

    // 16*96 planes * 112 output rows * 56 float4-groups per row
    const unsigned total_tasks = 16u * 96u * 112u * GW;  // 9,633,792
    const int threads = 256;                             // 8 wave32s per block
    const int blocks  = (int)((total_tasks + 255u) / 256u);  // 37,632

    maxpool2x2_s2_kernel<<<blocks, threads, 0, stream>>>(x, y, total_tasks);
}